// linear_sti_30915174596798
// MI455X (gfx1250) — compile-verified
//
#include <hip/hip_runtime.h>
#include <math.h>

// MI455X / gfx1250: wave32, WMMA fp32 16x16x4 matrix pipe + double-buffered
// async global->LDS copies (ASYNCcnt split issue/wait).

typedef __attribute__((ext_vector_type(2))) float v2f;
typedef __attribute__((ext_vector_type(4))) float v4f;
typedef __attribute__((ext_vector_type(8))) float v8f;

#define DD      2048   // d
#define MW      1024   // m (window, padded; real window = 1023)
#define NSTEPS  4096
#define NB      64

#define USE_ASYNC_LDS 1

// LDS byte address of a __shared__ object: low 32 bits of the flat address.
__device__ __forceinline__ unsigned lds_off_of(const void* p)
{
    return (unsigned)(unsigned long long)p;
}

// CDNA5 async copy: global -> LDS, 128 bits per lane, tracked by ASYNCcnt.
__device__ __forceinline__ void async_load_b128(void* lds_dst, const void* gsrc)
{
    unsigned l = lds_off_of(lds_dst);
    unsigned long long g = (unsigned long long)gsrc;
    asm volatile("global_load_async_to_lds_b128 %0, %1, off"
                 :: "v"(l), "v"(g) : "memory");
}
__device__ __forceinline__ void wait_async0()
{
    asm volatile("s_wait_asynccnt 0x0" ::: "memory");
}

// ---------------------------------------------------------------------------
// Generic fp32 WMMA GEMM:  C += alpha * opA(A) * opB(B)
//   MODE 0 (NN): A is MxK row-major (lda), B is KxN row-major (ldb)
//   MODE 1 (NT): A is MxK row-major (lda), B passed as Bt: N rows x K cols (ldb)
//   MODE 2 (TN): A passed as K rows x M cols (lda), B is KxN row-major (ldb)
//   MODE 3 (Hankel-NT): A is a 1D sequence s, A[m,k] = s[m0+m+k0+k]; B as NT.
// Block tile: 64(M) x 64(N); K-chunk 16; 256 threads = 8 waves in 4x2 grid,
// each wave owns two 16x16 C tiles (shared A fragment).
// Double-buffered LDS: chunk i+1's async copies overlap chunk i's WMMAs.
// Requires: M % 64 == 0, N % 64 == 0, K % 16 == 0 (all call sites satisfy).
// ---------------------------------------------------------------------------
template <int MODE>
__global__ __launch_bounds__(256)
void wmma_gemm(const float* __restrict__ A, int lda,
               const float* __restrict__ B, int ldb,
               float* __restrict__ C, int ldc,
               int K, float alpha)
{
    // Row stride 20 floats: keeps b128 LDS stores 16B-aligned and fragment
    // reads bank-conflict-free (20*m mod 64 distinct for m = 0..15).
    __shared__ float As[2][64][20];
    __shared__ float Bs[2][64][20];
    __shared__ float Ah[2][80];       // MODE 3: Hankel window of s

    const int tid  = threadIdx.x;
    const int lane = tid & 31;
    const int wave = tid >> 5;
    const int wm   = wave >> 1;          // 0..3  (M direction)
    const int wn   = wave & 1;           // 0..1  (N direction)
    const int m0   = blockIdx.y * 64;
    const int n0   = blockIdx.x * 64;

    v8f acc0 = {0.f, 0.f, 0.f, 0.f, 0.f, 0.f, 0.f, 0.f};
    v8f acc1 = {0.f, 0.f, 0.f, 0.f, 0.f, 0.f, 0.f, 0.f};

    // Stage one 64x16 A tile + 64x16 B tile into LDS buffer `buf`.
    auto stage = [&](int buf, int k0c) {
        // ---- A tile ----
        if (MODE == 0 || MODE == 1) {
            // contiguous in K: one float4 per thread
            int m = tid >> 2, k4 = (tid & 3) << 2;
            const float* src = &A[(size_t)(m0 + m) * lda + (k0c + k4)];
#if USE_ASYNC_LDS
            async_load_b128(&As[buf][m][k4], src);
#else
            *(v4f*)&As[buf][m][k4] = *(const v4f*)src;
#endif
        } else if (MODE == 2) {
            // A physical K x M, contiguous in M: float4 load, scatter 4 rows
            int k = tid >> 4, m4 = (tid & 15) << 2;
            v4f av = *(const v4f*)&A[(size_t)(k0c + k) * lda + (m0 + m4)];
            As[buf][m4 + 0][k] = av.x; As[buf][m4 + 1][k] = av.y;
            As[buf][m4 + 2][k] = av.z; As[buf][m4 + 3][k] = av.w;
        } else {
            // MODE 3: 80-float window of s covers the whole 64x16 tile
            if (tid < 80) Ah[buf][tid] = A[(size_t)(m0 + k0c + tid)];
        }
        // ---- B tile (stored Bs[n][k]) ----
        if (MODE == 1 || MODE == 3) {
            int n = tid >> 2, k4 = (tid & 3) << 2;
            const float* src = &B[(size_t)(n0 + n) * ldb + (k0c + k4)];
#if USE_ASYNC_LDS
            async_load_b128(&Bs[buf][n][k4], src);
#else
            *(v4f*)&Bs[buf][n][k4] = *(const v4f*)src;
#endif
        } else {
            int k = tid >> 4, n4 = (tid & 15) << 2;
            v4f bv = *(const v4f*)&B[(size_t)(k0c + k) * ldb + (n0 + n4)];
            Bs[buf][n4 + 0][k] = bv.x; Bs[buf][n4 + 1][k] = bv.y;
            Bs[buf][n4 + 2][k] = bv.z; Bs[buf][n4 + 3][k] = bv.w;
        }
    };

    stage(0, 0);   // prologue

    const int mrow = (wm << 4) + (lane & 15);
    const int nrow = (wn << 4) + (lane & 15);
    const int kb   = (lane >> 4) << 1;

    for (int k0 = 0; k0 < K; k0 += 16) {
        const int cur = (k0 >> 4) & 1;
#if USE_ASYNC_LDS
        if (MODE == 0 || MODE == 1 || MODE == 3) wait_async0();
#endif
        __syncthreads();                 // cur buffer ready; prev reads done
        if (k0 + 16 < K) stage(cur ^ 1, k0 + 16);   // overlap with compute

        // A frag (16x4 f32): lanes 0-15 -> K 0/1, lanes 16-31 -> K 2/3; M = lane&15
        // B frag (4x16 f32): mirrored; N = lane&15
        #pragma unroll
        for (int kk = 0; kk < 16; kk += 4) {
            v2f a, b0, b1;
            if (MODE == 3) {
                a.x = Ah[cur][mrow + kk + kb];
                a.y = Ah[cur][mrow + kk + kb + 1];
            } else {
                a.x = As[cur][mrow][kk + kb];
                a.y = As[cur][mrow][kk + kb + 1];
            }
            b0.x = Bs[cur][nrow][kk + kb];
            b0.y = Bs[cur][nrow][kk + kb + 1];
            b1.x = Bs[cur][nrow + 32][kk + kb];
            b1.y = Bs[cur][nrow + 32][kk + kb + 1];
            acc0 = __builtin_amdgcn_wmma_f32_16x16x4_f32(
                false, a, false, b0, (short)0, acc0, false, false);
            acc1 = __builtin_amdgcn_wmma_f32_16x16x4_f32(
                false, a, false, b1, (short)0, acc1, false, false);
        }
    }

    // Epilogue: C += alpha * acc.
    // C/D layout: VGPR r holds M = r (lanes 0-15) / M = r+8 (lanes 16-31), N = lane&15.
    const int col0  = n0 + (wn << 4) + (lane & 15);
    const int rbase = m0 + (wm << 4) + ((lane >> 4) << 3);
    #pragma unroll
    for (int r = 0; r < 8; ++r) {
        size_t idx = (size_t)(rbase + r) * ldc + col0;
        C[idx] += alpha * acc0[r];
        C[idx + 32] += alpha * acc1[r];
    }
}

// ---------------------------------------------------------------------------
// Z = copy of x (RHS = [Xh | x_last]);  Xp = x with last column zeroed.
// ---------------------------------------------------------------------------
__global__ void init_copies(const float* __restrict__ x,
                            float* __restrict__ Xp, float* __restrict__ Z)
{
    size_t i = (size_t)blockIdx.x * blockDim.x + threadIdx.x;
    if (i < (size_t)DD * MW) {
        float v = x[i];
        Z[i]  = v;
        Xp[i] = ((i & (size_t)(MW - 1)) == (size_t)(MW - 1)) ? 0.f : v;
    }
}

__global__ void add_identity(float* __restrict__ XX)
{
    int i = blockIdx.x * blockDim.x + threadIdx.x;
    if (i < DD) XX[(size_t)i * DD + i] += 1.0f;
}

__global__ void zero_zcol(float* __restrict__ Z)
{
    int i = blockIdx.x * blockDim.x + threadIdx.x;
    if (i < DD) Z[(size_t)i * MW + (MW - 1)] = 0.f;
}

// ---------------------------------------------------------------------------
// Cholesky: factor 64x64 diagonal block in-place (one block, 64 threads).
// ---------------------------------------------------------------------------
__global__ void potrf_diag(float* __restrict__ XX, int kb)
{
    __shared__ float Ld[64][65];
    const int tid = threadIdx.x;      // 64 threads
    float* Ablk = XX + (size_t)kb * NB * DD + kb * NB;
    for (int i = tid; i < 64 * 64; i += 64)
        Ld[i >> 6][i & 63] = Ablk[(size_t)(i >> 6) * DD + (i & 63)];
    __syncthreads();
    for (int j = 0; j < 64; ++j) {
        if (tid == j) Ld[j][j] = sqrtf(Ld[j][j]);
        __syncthreads();
        if (tid > j) Ld[tid][j] /= Ld[j][j];
        __syncthreads();
        if (tid > j) {
            float ljt = Ld[tid][j];
            for (int c = j + 1; c <= tid; ++c) Ld[tid][c] -= ljt * Ld[c][j];
        }
        __syncthreads();
    }
    for (int i = tid; i < 64 * 64; i += 64)
        Ablk[(size_t)(i >> 6) * DD + (i & 63)] = Ld[i >> 6][i & 63];
}

// Panel: P <- P * inv(L_kk^T); one thread per panel row (R rows below diag).
__global__ __launch_bounds__(256)
void trsm_panel(float* __restrict__ XX, int kb, int R)
{
    __shared__ float Lkk[64][65];
    const int tid = threadIdx.x;
    const float* Dblk = XX + (size_t)kb * NB * DD + kb * NB;
    for (int i = tid; i < 64 * 64; i += 256)
        Lkk[i >> 6][i & 63] = Dblk[(size_t)(i >> 6) * DD + (i & 63)];
    __syncthreads();
    int r = blockIdx.x * 256 + tid;
    if (r >= R) return;
    float* Prow = XX + (size_t)((kb + 1) * NB + r) * DD + kb * NB;
    float pr[64];
    for (int j = 0; j < 64; ++j) {
        float v = Prow[j];
        for (int i = 0; i < j; ++i) v -= pr[i] * Lkk[j][i];
        pr[j] = v / Lkk[j][j];
    }
    for (int j = 0; j < 64; ++j) Prow[j] = pr[j];
}

// Forward diag solve: L_kk * Z_k = Z_k; one thread per RHS column (1024 cols).
__global__ __launch_bounds__(256)
void trsm_diag_lower(const float* __restrict__ XX, float* __restrict__ Z, int kb)
{
    __shared__ float Lkk[64][65];
    const int tid = threadIdx.x;
    const float* Dblk = XX + (size_t)kb * NB * DD + kb * NB;
    for (int i = tid; i < 64 * 64; i += 256)
        Lkk[i >> 6][i & 63] = Dblk[(size_t)(i >> 6) * DD + (i & 63)];
    __syncthreads();
    int c = blockIdx.x * 256 + tid;
    float* Zb = Z + (size_t)kb * NB * MW + c;
    float zr[64];
    for (int j = 0; j < 64; ++j) {
        float v = Zb[(size_t)j * MW];
        for (int i = 0; i < j; ++i) v -= Lkk[j][i] * zr[i];
        zr[j] = v / Lkk[j][j];
    }
    for (int j = 0; j < 64; ++j) Zb[(size_t)j * MW] = zr[j];
}

// Backward diag solve: L_kk^T * Z_k = Z_k (reads L columns).
__global__ __launch_bounds__(256)
void trsm_diag_upperT(const float* __restrict__ XX, float* __restrict__ Z, int kb)
{
    __shared__ float Lkk[64][65];
    const int tid = threadIdx.x;
    const float* Dblk = XX + (size_t)kb * NB * DD + kb * NB;
    for (int i = tid; i < 64 * 64; i += 256)
        Lkk[i >> 6][i & 63] = Dblk[(size_t)(i >> 6) * DD + (i & 63)];
    __syncthreads();
    int c = blockIdx.x * 256 + tid;
    float* Zb = Z + (size_t)kb * NB * MW + c;
    float zr[64];
    for (int j = 63; j >= 0; --j) {
        float v = Zb[(size_t)j * MW];
        for (int i = j + 1; i < 64; ++i) v -= Lkk[i][j] * zr[i];
        zr[j] = v / Lkk[j][j];
    }
    for (int j = 0; j < 64; ++j) Zb[(size_t)j * MW] = zr[j];
}

// g[k] = sum_i Xp[i,k] * v[i], with v = Z[:, MW-1].  (Xp col 1023 is zero -> g[1023]=0.)
__global__ __launch_bounds__(256)
void compute_g(const float* __restrict__ Xp, const float* __restrict__ Z,
               float* __restrict__ g)
{
    __shared__ float red[256];
    const int k = blockIdx.x;
    float p = 0.f;
    for (int i = threadIdx.x; i < DD; i += 256)
        p += Xp[(size_t)i * MW + k] * Z[(size_t)i * MW + (MW - 1)];
    red[threadIdx.x] = p;
    __syncthreads();
    for (int off = 128; off > 0; off >>= 1) {
        if (threadIdx.x < off) red[threadIdx.x] += red[threadIdx.x + off];
        __syncthreads();
    }
    if (threadIdx.x == 0) g[k] = red[0];
}

// ---------------------------------------------------------------------------
// Sequential AR scan, single wave (32 lanes), zero barriers:
//   s[0..1022] = y[1..1023];  s[1023+t] = dot(s[t..t+1022], g)
// Each lane keeps its 32 filter taps in VGPRs; window values come from LDS
// (same-wave LDS ops are in-order, so lane 0's append is seen next step).
// Tail is exact: g[1023] == 0 and ss[1023+t] is pre-zeroed before first use.
// Also writes y_out = [y[0..1023], predictions] into d_out[0..5119].
// ---------------------------------------------------------------------------
__global__ __launch_bounds__(32)
void ar_scan(const float* __restrict__ y, const float* __restrict__ g,
             float* __restrict__ s_out, float* __restrict__ y_out)
{
    __shared__ float ss[1023 + NSTEPS + 1];
    const int lane = threadIdx.x;            // 0..31

    float gr[32];
    #pragma unroll
    for (int j = 0; j < 32; ++j) gr[j] = g[lane * 32 + j];   // g[1023] == 0

    for (int i = lane; i < MW - 1; i += 32) ss[i] = y[1 + i];
    for (int i = lane; i < NSTEPS + 1; i += 32) ss[1023 + i] = 0.f;
    for (int i = lane; i < MW; i += 32) y_out[i] = y[i];
    __syncthreads();

    const int base = lane * 32;
    for (int t = 0; t < NSTEPS; ++t) {
        float p = 0.f;
        #pragma unroll
        for (int j = 0; j < 32; ++j) p += ss[t + base + j] * gr[j];
        #pragma unroll
        for (int off = 16; off > 0; off >>= 1) p += __shfl_xor(p, off, 32);
        if (lane == 0) ss[1023 + t] = p;     // in-order LDS: visible next step
    }

    for (int i = lane; i < 1023 + NSTEPS; i += 32) s_out[i] = ss[i];
    if (lane == 0) s_out[1023 + NSTEPS] = 0.f;   // pad element 5119 (read, unused)
    for (int t = lane; t < NSTEPS; t += 32) y_out[MW + t] = ss[1023 + t];
}

// ---------------------------------------------------------------------------
// Host-side orchestration (all on `stream`; graph-capture safe).
// ---------------------------------------------------------------------------
extern "C" void kernel_launch(void* const* d_in, const int* in_sizes, int n_in,
                              void* d_out, int out_size, void* d_ws, size_t ws_size,
                              hipStream_t stream)
{
    (void)in_sizes; (void)n_in; (void)out_size; (void)ws_size;
    const float* x = (const float*)d_in[0];    // (2048, 1024)
    const float* y = (const float*)d_in[1];    // (5120,)
    float* out = (float*)d_out;                // [y_out(5120) | A(4096*2048)]

    float* ws   = (float*)d_ws;
    float* XX   = ws;                                   // 2048*2048
    float* Z    = XX + (size_t)DD * DD;                 // 2048*1024  (RHS -> inv(XX)*[Xh|x_last])
    float* Xp   = Z  + (size_t)DD * MW;                 // 2048*1024  (x with col 1023 zeroed)
    float* g    = Xp + (size_t)DD * MW;                 // 1024
    float* sbuf = g  + MW;                              // 5120

    // 1) copies: Z = x; Xp = x with last column zeroed
    init_copies<<<(DD * MW + 255) / 256, 256, 0, stream>>>(x, Xp, Z);

    // 2) XX = Xp * Xp^T + I   (WMMA NT GEMM, K padded to 1024 via zeroed column)
    hipMemsetAsync(XX, 0, (size_t)DD * DD * sizeof(float), stream);
    wmma_gemm<1><<<dim3(DD / 64, DD / 64), 256, 0, stream>>>(Xp, MW, Xp, MW, XX, DD, MW, 1.0f);
    add_identity<<<(DD + 255) / 256, 256, 0, stream>>>(XX);

    // 3) Blocked Cholesky XX = L L^T (in place, lower)
    for (int kb = 0; kb < DD / NB; ++kb) {
        potrf_diag<<<1, 64, 0, stream>>>(XX, kb);
        int R = DD - (kb + 1) * NB;
        if (R > 0) {
            trsm_panel<<<(R + 255) / 256, 256, 0, stream>>>(XX, kb, R);
            const float* P = XX + (size_t)(kb + 1) * NB * DD + kb * NB;
            float* Ct = XX + (size_t)(kb + 1) * NB * DD + (kb + 1) * NB;
            wmma_gemm<1><<<dim3(R / 64, R / 64), 256, 0, stream>>>(P, DD, P, DD, Ct, DD, NB, -1.0f);
        }
    }

    // 4) Forward solve: L * Z = Z
    for (int kb = 0; kb < DD / NB; ++kb) {
        trsm_diag_lower<<<MW / 256, 256, 0, stream>>>(XX, Z, kb);
        int R = DD - (kb + 1) * NB;
        if (R > 0) {
            const float* P  = XX + (size_t)(kb + 1) * NB * DD + kb * NB;
            const float* Zk = Z + (size_t)kb * NB * MW;
            float* Zr = Z + (size_t)(kb + 1) * NB * MW;
            wmma_gemm<0><<<dim3(MW / 64, R / 64), 256, 0, stream>>>(P, DD, Zk, MW, Zr, MW, NB, -1.0f);
        }
    }

    // 5) Backward solve: L^T * Z = Z  (TN update, then diag solve, bottom-up)
    for (int kb = DD / NB - 1; kb >= 0; --kb) {
        int R = DD - (kb + 1) * NB;
        if (R > 0) {
            const float* P  = XX + (size_t)(kb + 1) * NB * DD + kb * NB;  // R x 64, used transposed
            const float* Zb = Z + (size_t)(kb + 1) * NB * MW;
            float* Zk = Z + (size_t)kb * NB * MW;
            wmma_gemm<2><<<dim3(MW / 64, 1), 256, 0, stream>>>(P, DD, Zb, MW, Zk, MW, R, -1.0f);
        }
        trsm_diag_upperT<<<MW / 256, 256, 0, stream>>>(XX, Z, kb);
    }

    // 6) g = Xh^T v  (v = Z[:, 1023]); then zero Z's last column for the padded-K GEMM
    compute_g<<<MW, 256, 0, stream>>>(Xp, Z, g);
    zero_zcol<<<(DD + 255) / 256, 256, 0, stream>>>(Z);

    // 7) AR recurrence -> s buffer and y_out (single wave, barrier-free)
    ar_scan<<<1, 32, 0, stream>>>(y, g, sbuf, out);

    // 8) A = S * Z^T  (Hankel S read straight from the 1D s buffer, MODE 3)
    hipMemsetAsync(out + 5120, 0, (size_t)NSTEPS * DD * sizeof(float), stream);
    wmma_gemm<3><<<dim3(DD / 64, NSTEPS / 64), 256, 0, stream>>>(sbuf, 1, Z, MW, out + 5120, DD, MW, 1.0f);
}